// PSROIAlignExample_85272280694830
// MI455X (gfx1250) — compile-verified
//
#include <hip/hip_runtime.h>
#include <hip/hip_bf16.h>

// PS-ROI Align, position-sensitive, torchvision "aligned" variant.
// feat: (B=4, C=490, H=100, W=100) fp32, rois: (N, 5) fp32
// out:  (N, PD=10, RS=7, RS=7) fp32
//
// One block per ROI: blockIdx.x is the ROI index, so the 5-float ROI
// descriptor is a wave-uniform load -> scalarized to s_load_b128/s_load_b32
// (constant cache, KMcnt), and all ROI->bin math runs on the SALU.
// Each of the 490 active threads produces one output element with 16
// independent global_load_b32 gathers issued back-to-back for MLP.
// feat (78.4 MB) is L2-resident on MI455X (192 MB L2).

#define RS        7
#define SR        2
#define PD        10
#define FEAT_H    100
#define FEAT_W    100
#define FEAT_C    (PD * RS * RS)   // 490
#define SSCALE    (1.0f / 16.0f)

__global__ __launch_bounds__(512, 4)
void psroi_align_kernel(const float* __restrict__ feat,
                        const float* __restrict__ rois,
                        float* __restrict__ out)
{
    const int tid = threadIdx.x;
    if (tid >= FEAT_C) return;          // 490 active lanes of 512

    const int n = blockIdx.x;           // ROI index: uniform -> SGPR

    // ROI descriptor: uniform address -> scalar loads via constant cache.
    const float* __restrict__ r = rois + (size_t)n * 5;
    const int   b  = (int)r[0];
    const float sw = r[1] * SSCALE - 0.5f;
    const float sh = r[2] * SSCALE - 0.5f;
    const float ew = r[3] * SSCALE - 0.5f;
    const float eh = r[4] * SSCALE - 0.5f;

    const float bin_w = (ew - sw) * (1.0f / RS);   // SALU
    const float bin_h = (eh - sh) * (1.0f / RS);

    // Decompose tid -> (p, ph, pw); pw fastest -> coalesced stores.
    const int pw = tid % RS;
    const int ph = (tid / RS) % RS;
    const int p  = tid / (RS * RS);

    // Position-sensitive channel c == tid by construction: (p*7+ph)*7+pw.
    const float* __restrict__ fp =
        feat + ((size_t)b * FEAT_C + (size_t)tid) * (FEAT_H * FEAT_W);

    // ---- Per-sample 1D interpolation data (y depends on iy, x on ix) -------
    int   y0i[SR], y1i[SR], x0i[SR], x1i[SR];
    float ly[SR], hy[SR], lx[SR], hx[SR];
    bool  vy[SR], vx[SR];

#pragma unroll
    for (int s = 0; s < SR; ++s) {
        const float g = ((float)s + 0.5f) * (1.0f / SR);
        // y sample
        float y = sh + ((float)ph + g) * bin_h;
        vy[s] = (y >= -1.0f) && (y <= (float)FEAT_H);
        y = fmaxf(y, 0.0f);
        int y0 = min((int)floorf(y), FEAT_H - 1);
        y0i[s] = y0;
        y1i[s] = min(y0 + 1, FEAT_H - 1);
        const float yc = (y0 >= FEAT_H - 1) ? (float)y0 : y;
        ly[s] = yc - (float)y0;
        hy[s] = 1.0f - ly[s];
        // x sample
        float x = sw + ((float)pw + g) * bin_w;
        vx[s] = (x >= -1.0f) && (x <= (float)FEAT_W);
        x = fmaxf(x, 0.0f);
        int x0 = min((int)floorf(x), FEAT_W - 1);
        x0i[s] = x0;
        x1i[s] = min(x0 + 1, FEAT_W - 1);
        const float xc = (x0 >= FEAT_W - 1) ? (float)x0 : x;
        lx[s] = xc - (float)x0;
        hx[s] = 1.0f - lx[s];
    }

    // Warm the first cacheline while remaining address math retires.
    __builtin_prefetch(&fp[y0i[0] * FEAT_W + x0i[0]], 0, 3);

    // ---- Issue all 16 gathers, then combine (loads hoisted for MLP) --------
    float f00[SR][SR], f01[SR][SR], f10[SR][SR], f11[SR][SR];
#pragma unroll
    for (int sy = 0; sy < SR; ++sy) {
        const int r0 = y0i[sy] * FEAT_W;
        const int r1 = y1i[sy] * FEAT_W;
#pragma unroll
        for (int sx = 0; sx < SR; ++sx) {
            f00[sy][sx] = fp[r0 + x0i[sx]];
            f01[sy][sx] = fp[r0 + x1i[sx]];
            f10[sy][sx] = fp[r1 + x0i[sx]];
            f11[sy][sx] = fp[r1 + x1i[sx]];
        }
    }

    float acc = 0.0f;
#pragma unroll
    for (int sy = 0; sy < SR; ++sy) {
#pragma unroll
        for (int sx = 0; sx < SR; ++sx) {
            float v = hy[sy] * hx[sx] * f00[sy][sx];
            v = fmaf(hy[sy] * lx[sx], f01[sy][sx], v);
            v = fmaf(ly[sy] * hx[sx], f10[sy][sx], v);
            v = fmaf(ly[sy] * lx[sx], f11[sy][sx], v);
            if (vy[sy] && vx[sx]) acc += v;
        }
    }

    out[(size_t)n * FEAT_C + tid] = acc * (1.0f / (SR * SR));
}

extern "C" void kernel_launch(void* const* d_in, const int* in_sizes, int n_in,
                              void* d_out, int out_size, void* d_ws, size_t ws_size,
                              hipStream_t stream)
{
    const float* feat = (const float*)d_in[0];
    const float* rois = (const float*)d_in[1];
    float* out = (float*)d_out;

    const int n_rois = in_sizes[1] / 5;   // (N, 5)
    const int block  = 512;               // 16 wave32 waves; 490 active lanes
    const int grid   = n_rois;            // one block per ROI

    psroi_align_kernel<<<grid, block, 0, stream>>>(feat, rois, out);
}